// FOTSLoss_33423435497883
// MI455X (gfx1250) — compile-verified
//
#include <hip/hip_runtime.h>
#include <hip/hip_bf16.h>
#include <stdint.h>

#define NBATCH   64
#define NPIX     (512 * 512)       // 262144 elements per batch
#define TB       1024              // threads per block (32 wave32 waves)
#define K_HARD   512
#define K_RAND   512
#define FEPS     1e-7f
#define TILE     4096              // floats per TDM transfer (16 KB), x2 buffers
#define NTILES   (NPIX / TILE)     // 64
#define EPT      (TILE / TB)       // 4 elements / thread / tile
#define EPTH     (NPIX / TB)       // 256 elements / thread total

typedef __attribute__((ext_vector_type(4))) unsigned int v4u;
typedef __attribute__((ext_vector_type(8))) int          v8i;
typedef __attribute__((ext_vector_type(4))) int          v4i;

// ---------------------------------------------------------------------------
// Tensor Data Mover: DMA `nelem` f32 from global `gsrc` into LDS byte address
// `lds_addr`. 1-D tensor, 1-D tile. Descriptor per CDNA5 ISA §8 (CDNA5 ISA).
// clang-23 toolchain: 6-arg builtin (extra int32x8 group, zero for <=2-D).
// ---------------------------------------------------------------------------
__device__ __forceinline__ void tdm_load_tile_f32(const float* gsrc,
                                                  uint32_t lds_addr,
                                                  uint32_t nelem) {
  uint64_t ga = (uint64_t)(uintptr_t)gsrc;
  v4u g0;
  g0.x = 1u;                                        // count=1 (valid), no gather
  g0.y = lds_addr;                                  // lds_addr [63:32]
  g0.z = (uint32_t)(ga & 0xFFFFFFFFu);              // global_addr [95:64]
  g0.w = (uint32_t)((ga >> 32) & 0x01FFFFFFu)       // global_addr [120:96]
         | (2u << 30);                              // type=2 ("image")
  v8i g1;
  g1[0] = (int)(2u << 16);                          // data_size=2 (4 bytes), mask=0
  g1[1] = (int)((nelem & 0xFFFFu) << 16);           // tensor_dim0[15:0] @ bits 63:48
  g1[2] = (int)(((nelem >> 16) & 0xFFFFu)           // tensor_dim0[31:16]
                | (1u << 16));                      // tensor_dim1 = 1
  g1[3] = (int)((nelem & 0xFFFFu) << 16);           // tile_dim0 @ bits 127:112
  g1[4] = 0;                                        // tile_dim1=0, tile_dim2=0 (unused)
  g1[5] = (int)nelem;                               // tensor_dim0_stride[31:0]
  g1[6] = 0;                                        // stride hi / dim1_stride lo
  g1[7] = 0;
  v4i gz4  = {0, 0, 0, 0};                          // groups 2/3 unused (<=2D)
  v8i gz8  = {0, 0, 0, 0, 0, 0, 0, 0};              // extra group (clang-23 form)
  __builtin_amdgcn_tensor_load_to_lds(g0, g1, gz4, gz4, gz8, 0);
}

// Deterministic block tree-reduction (no float atomics).
__device__ __forceinline__ float block_sum(float v, float* red, int tid) {
  red[tid] = v;
  __syncthreads();
  for (int s = TB / 2; s > 0; s >>= 1) {
    if (tid < s) red[tid] += red[tid + s];
    __syncthreads();
  }
  float r = red[0];
  __syncthreads();
  return r;
}

__global__ __launch_bounds__(TB)
void fots_ohem_loss_kernel(const unsigned char* __restrict__ pos_all,
                           const float* __restrict__ conf_all,
                           const float* __restrict__ u_all,
                           float* __restrict__ out) {
  __shared__ uint32_t s_hist[2048];      // radix-select histogram  (8 KB)
  __shared__ float    s_red[TB];         // reduction scratch       (4 KB)
  __shared__ uint32_t s_b[8];            // scan broadcast
  __shared__ float    s_tile[2][TILE];   // TDM ping-pong tiles     (32 KB)

  const int tid = threadIdx.x;
  const int b   = blockIdx.x;
  const unsigned char* pos = pos_all + (size_t)b * NPIX;
  const float* conf        = conf_all + (size_t)b * NPIX;
  const float* uu          = u_all + (size_t)b * NPIX;

  const uint32_t tile_lds0 = (uint32_t)(uintptr_t)(void*)&s_tile[0][0];
  const uint32_t tile_lds1 = (uint32_t)(uintptr_t)(void*)&s_tile[1][0];

  // ---------------- helpers ----------------
  auto scan_hist = [&](int nbins, uint32_t kneed) {
    if (tid == 0) {
      uint32_t run = 0;
      int sel = -1;
      uint32_t at = 0;
      for (int bb = nbins - 1; bb >= 0; --bb) {
        uint32_t h = s_hist[bb];
        if (run + h >= kneed) { sel = bb; at = h; break; }
        run += h;
      }
      s_b[0] = (uint32_t)sel;           // selected bin (0xFFFFFFFF if exhausted)
      s_b[1] = run;                     // count strictly above selected bin
      s_b[2] = at;                      // count inside selected bin
      s_b[3] = (sel < 0) ? 1u : 0u;     // exhausted: fewer candidates than kneed
    }
    __syncthreads();
  };
  auto zero_hist = [&]() {
    for (int i = tid; i < 2048; i += TB) s_hist[i] = 0u;
    __syncthreads();
  };
  // Ping-pong TDM pipeline driver (wave 0 only): issue tile t+1, then wait
  // until tile t is resident. Same-wave TDM ops retire in order, so
  // s_wait_tensorcnt(1) == "tile t complete, tile t+1 may still fly".
  auto tdm_pipeline_step = [&](const float* src, int t) {
    if (tid < 32) {
      if (t + 1 < NTILES) {
        tdm_load_tile_f32(src + (size_t)(t + 1) * TILE,
                          ((t + 1) & 1) ? tile_lds1 : tile_lds0, TILE);
        __builtin_amdgcn_s_wait_tensorcnt(1);
      } else {
        __builtin_amdgcn_s_wait_tensorcnt(0);
      }
    }
    __syncthreads();                    // tile t visible to all waves
  };

  // =========== C-select level 0 (top 11 bits) — TDM-staged conf ===========
  zero_hist();
  if (tid < 32) tdm_load_tile_f32(conf, tile_lds0, TILE);   // prime tile 0
  for (int t = 0; t < NTILES; ++t) {
    tdm_pipeline_step(conf, t);
    const float* tl = s_tile[t & 1];
#pragma unroll
    for (int k = 0; k < EPT; ++k) {
      int li  = tid + k * TB;
      int idx = t * TILE + li;
      float c = fminf(fmaxf(tl[li], FEPS), 1.0f - FEPS);
      if (pos[idx] == 0)
        atomicAdd(&s_hist[__float_as_uint(c) >> 21], 1u);
    }
    __syncthreads();                    // tile consumed, safe to overwrite
  }
  scan_hist(2048, K_HARD);
  const uint32_t hard_all = s_b[3];     // total negatives < K_HARD
  uint32_t pfx    = s_b[0];
  uint32_t cnt_gt = s_b[1];
  uint32_t krem   = K_HARD - cnt_gt;
  uint32_t Tbits  = 0, picked_at = 0;

  if (!hard_all) {
    // ------- level 1 (bits 20:10) — conf re-read hits L2 (192 MB) -------
    zero_hist();
    for (int j = 0; j < EPTH; ++j) {
      int idx = tid + j * TB;
      float c = fminf(fmaxf(conf[idx], FEPS), 1.0f - FEPS);
      uint32_t cb = __float_as_uint(c);
      if (pos[idx] == 0 && (cb >> 21) == pfx)
        atomicAdd(&s_hist[(cb >> 10) & 0x7FFu], 1u);
    }
    __syncthreads();
    scan_hist(2048, krem);
    pfx    = (pfx << 11) | s_b[0];
    cnt_gt += s_b[1];
    krem   -= s_b[1];
    // ------- level 2 (bits 9:0) -------
    zero_hist();
    for (int j = 0; j < EPTH; ++j) {
      int idx = tid + j * TB;
      float c = fminf(fmaxf(conf[idx], FEPS), 1.0f - FEPS);
      uint32_t cb = __float_as_uint(c);
      if (pos[idx] == 0 && (cb >> 10) == pfx)
        atomicAdd(&s_hist[cb & 0x3FFu], 1u);
    }
    __syncthreads();
    scan_hist(1024, krem);
    Tbits     = (pfx << 10) | s_b[0];
    cnt_gt   += s_b[1];
    picked_at = krem - s_b[1];          // == K_HARD - cnt_gt, ties at T
  }

  // =========== pos_loss + hard_loss sweep ===========
  float acc = 0.0f;
  for (int j = 0; j < EPTH; ++j) {
    int idx = tid + j * TB;
    float c = fminf(fmaxf(conf[idx], FEPS), 1.0f - FEPS);
    if (pos[idx] != 0) {
      acc += -logf(c);
    } else {
      uint32_t cb = __float_as_uint(c);
      if (hard_all || cb > Tbits) acc += -log1pf(-c);
    }
  }
  float total = block_sum(acc, s_red, tid);
  if (!hard_all)
    total += (float)picked_at * (-log1pf(-__uint_as_float(Tbits)));

  // =========== random-negative select on u over remaining ===========
  uint32_t u_all_flag = 1, Ubits = 0, upicked = 0, ucnt_at = 1;
  if (!hard_all) {
    // ------- level 0 — TDM-staged u keys -------
    zero_hist();
    if (tid < 32) tdm_load_tile_f32(uu, tile_lds0, TILE);   // prime tile 0
    for (int t = 0; t < NTILES; ++t) {
      tdm_pipeline_step(uu, t);
      const float* tl = s_tile[t & 1];
#pragma unroll
      for (int k = 0; k < EPT; ++k) {
        int li  = tid + k * TB;
        int idx = t * TILE + li;
        float c = fminf(fmaxf(conf[idx], FEPS), 1.0f - FEPS);
        if (pos[idx] == 0 && __float_as_uint(c) < Tbits)
          atomicAdd(&s_hist[__float_as_uint(tl[li]) >> 21], 1u);
      }
      __syncthreads();
    }
    scan_hist(2048, K_RAND);
    u_all_flag       = s_b[3];          // remaining < K_RAND: take all
    uint32_t upfx    = s_b[0];
    uint32_t ucnt_gt = s_b[1];
    uint32_t ukrem   = K_RAND - ucnt_gt;
    if (!u_all_flag) {
      // ------- level 1 -------
      zero_hist();
      for (int j = 0; j < EPTH; ++j) {
        int idx = tid + j * TB;
        float c = fminf(fmaxf(conf[idx], FEPS), 1.0f - FEPS);
        if (pos[idx] == 0 && __float_as_uint(c) < Tbits) {
          uint32_t ub = __float_as_uint(uu[idx]);
          if ((ub >> 21) == upfx) atomicAdd(&s_hist[(ub >> 10) & 0x7FFu], 1u);
        }
      }
      __syncthreads();
      scan_hist(2048, ukrem);
      upfx   = (upfx << 11) | s_b[0];
      ukrem -= s_b[1];
      // ------- level 2 -------
      zero_hist();
      for (int j = 0; j < EPTH; ++j) {
        int idx = tid + j * TB;
        float c = fminf(fmaxf(conf[idx], FEPS), 1.0f - FEPS);
        if (pos[idx] == 0 && __float_as_uint(c) < Tbits) {
          uint32_t ub = __float_as_uint(uu[idx]);
          if ((ub >> 10) == upfx) atomicAdd(&s_hist[ub & 0x3FFu], 1u);
        }
      }
      __syncthreads();
      scan_hist(1024, ukrem);
      Ubits   = (upfx << 10) | s_b[0];
      upicked = ukrem - s_b[1];         // picks among u-ties
      ucnt_at = (s_b[2] > 0) ? s_b[2] : 1u;
    }
    // ------- rand_loss sweep -------
    float racc = 0.0f, tacc = 0.0f;
    for (int j = 0; j < EPTH; ++j) {
      int idx = tid + j * TB;
      float c = fminf(fmaxf(conf[idx], FEPS), 1.0f - FEPS);
      if (pos[idx] == 0 && __float_as_uint(c) < Tbits) {
        float lv = -log1pf(-c);
        if (u_all_flag) {
          racc += lv;
        } else {
          uint32_t ub = __float_as_uint(uu[idx]);
          if (ub > Ubits)       racc += lv;
          else if (ub == Ubits) tacc += lv;
        }
      }
    }
    float sr = block_sum(racc, s_red, tid);
    float st = block_sum(tacc, s_red, tid);
    total += sr;
    if (!u_all_flag)
      total += st * ((float)upicked / (float)ucnt_at);
  }

  if (tid == 0) out[b] = total;
}

extern "C" void kernel_launch(void* const* d_in, const int* in_sizes, int n_in,
                              void* d_out, int out_size, void* d_ws, size_t ws_size,
                              hipStream_t stream) {
  (void)in_sizes; (void)n_in; (void)d_ws; (void)ws_size; (void)out_size;
  const unsigned char* pos = (const unsigned char*)d_in[0];  // bool (b,h,w)
  const float* conf        = (const float*)d_in[1];          // f32 (b,h,w,1)
  const float* u           = (const float*)d_in[2];          // f32 (b,h,w)
  float* out               = (float*)d_out;                  // f32 (b,)
  fots_ohem_loss_kernel<<<dim3(NBATCH), dim3(TB), 0, stream>>>(pos, conf, u, out);
}